// MultiGranularityEncoder_13915694039213
// MI455X (gfx1250) — compile-verified
//
#include <hip/hip_runtime.h>
#include <hip/hip_bf16.h>

typedef float v2f __attribute__((ext_vector_type(2)));
typedef float v8f __attribute__((ext_vector_type(8)));

#define EMB 128
#define NEG_SLOPE 0.2f

// ---------------------------------------------------------------------------
// helpers
// ---------------------------------------------------------------------------
__device__ __forceinline__ void atomicMaxF(float* addr, float val) {
    // classic IEEE-float ordered atomic max (works with -inf init)
    if (val >= 0.0f) atomicMax((int*)addr, __float_as_int(val));
    else             atomicMin((unsigned int*)addr, __float_as_uint(val));
}

__device__ __forceinline__ float lrelu(float x) {
    return x > 0.0f ? x : NEG_SLOPE * x;
}

// ---------------------------------------------------------------------------
// fill
// ---------------------------------------------------------------------------
__global__ void k_fill(float* __restrict__ p, float v, int n) {
    int t = blockIdx.x * 256 + threadIdx.x;
    if (t < n) p[t] = v;
}

// out1[i,c] = emb[ids[i],c] (residual of g0) + b0[c] + b4[c]
__global__ void k_init_out1(const float* __restrict__ emb, const int* __restrict__ ids,
                            const float* __restrict__ convb, float* __restrict__ out1, int n) {
    int t = blockIdx.x * 256 + threadIdx.x;
    if (t >= n) return;
    int node = t >> 7, c = t & 127;
    out1[t] = emb[(long long)ids[node] * EMB + c] + convb[0 * EMB + c] + convb[4 * EMB + c];
}

// out2[i,c] = b1[c] + b3[c] + b6[c]
__global__ void k_init_out2(const float* __restrict__ convb, float* __restrict__ out2, int n) {
    int t = blockIdx.x * 256 + threadIdx.x;
    if (t >= n) return;
    int c = t & 127;
    out2[t] = convb[1 * EMB + c] + convb[3 * EMB + c] + convb[6 * EMB + c];
}

// ---------------------------------------------------------------------------
// WMMA f32 16x16x4 GEMM:  Z0 = gather(emb, ids) @ W0 ; Z3 = gather @ W3
// One wave per 16-row M tile.  A frags register-resident across K=128 and
// reused for both weight matrices.  B frags for a whole N-tile are batched
// into registers first so the backend emits one load clause + one wait
// followed by 32 back-to-back accumulating WMMAs (no per-WMMA stalls).
// Layouts (ISA 7.12.2, wave32):
//   A 16x4 : lane L -> m=L&15, khalf=L>>4 ; v0=A[m][k0+2k],v1=A[m][k0+2k+1]
//   B 4x16 : lane L -> n=L&15, khalf      ; v0=B[k0+2k][n],v1=B[k0+2k+1][n]
//   C 16x16: vgpr i, lane L -> row=i+8*khalf, col=L&15
// ---------------------------------------------------------------------------
__global__ void __launch_bounds__(256)
k_gemm_gather_dual(const float* __restrict__ emb, const int* __restrict__ ids,
                   const float* __restrict__ W0, const float* __restrict__ W3,
                   float* __restrict__ Z0, float* __restrict__ Z3, int numTiles) {
    int wave = threadIdx.x >> 5;
    int lane = threadIdx.x & 31;
    int tile = blockIdx.x * 8 + wave;
    if (tile >= numTiles) return;
    int khalf = lane >> 4;
    int mloc  = lane & 15;
    int r0 = tile * 16;
    const float* arow = emb + (long long)ids[r0 + mloc] * EMB;

    v2f a[32];
#pragma unroll
    for (int ks = 0; ks < 32; ++ks) {
        int col = ks * 4 + khalf * 2;
        a[ks][0] = arow[col];
        a[ks][1] = arow[col + 1];
    }

#pragma unroll
    for (int w = 0; w < 2; ++w) {
        const float* Wp = w ? W3 : W0;
        float*       Zp = w ? Z3 : Z0;
#pragma unroll
        for (int nt = 0; nt < 8; ++nt) {
            int ncol = nt * 16 + mloc;
            // batch-load all B fragments for this N-tile (one clause, one wait)
            v2f b[32];
#pragma unroll
            for (int ks = 0; ks < 32; ++ks) {
                int kk = ks * 4 + khalf * 2;
                b[ks][0] = Wp[kk * EMB + ncol];
                b[ks][1] = Wp[(kk + 1) * EMB + ncol];
            }
            v8f c = {};
#pragma unroll
            for (int ks = 0; ks < 32; ++ks) {
                c = __builtin_amdgcn_wmma_f32_16x16x4_f32(
                        false, a[ks], false, b[ks], (short)0, c, false, false);
            }
#pragma unroll
            for (int i = 0; i < 8; ++i) {
                int row = r0 + i + khalf * 8;
                Zp[(long long)row * EMB + ncol] = c[i];
            }
        }
    }
}

// ---------------------------------------------------------------------------
// per-(node,head) attention scores: el0=sum(Z0*al0), er0=sum(Z0*ar0), el3=sum(Z3*al3)
// ---------------------------------------------------------------------------
__global__ void k_scores(const float* __restrict__ Z0, const float* __restrict__ Z3,
                         const float* __restrict__ attn_l, const float* __restrict__ attn_r,
                         float* __restrict__ el0, float* __restrict__ er0,
                         float* __restrict__ el3, int n /* N1*8 */) {
    int t = blockIdx.x * 256 + threadIdx.x;
    if (t >= n) return;
    int node = t >> 3, h = t & 7;
    const float* z0  = Z0 + (long long)node * EMB + h * 16;
    const float* z3  = Z3 + (long long)node * EMB + h * 16;
    const float* al0 = attn_l + 0 * EMB + h * 16;
    const float* ar0 = attn_r + 0 * EMB + h * 16;
    const float* al3 = attn_l + 3 * EMB + h * 16;
    float s0 = 0.f, s1 = 0.f, s3 = 0.f;
#pragma unroll
    for (int d = 0; d < 16; ++d) {
        s0 += z0[d] * al0[d];
        s1 += z0[d] * ar0[d];
        s3 += z3[d] * al3[d];
    }
    el0[t] = s0; er0[t] = s1; el3[t] = s3;
}

// ---------------------------------------------------------------------------
// edge softmax passes (er may be null => treated as 0, for the up1 graph)
// ---------------------------------------------------------------------------
__global__ void k_edge_max(const int* __restrict__ src, const int* __restrict__ dst,
                           const float* __restrict__ el, const float* __restrict__ er,
                           float* __restrict__ m, int E8) {
    int t = blockIdx.x * 256 + threadIdx.x;
    if (t >= E8) return;
    int e = t >> 3, h = t & 7;
    int d = dst[e];
    float v = el[src[e] * 8 + h] + (er ? er[d * 8 + h] : 0.0f);
    atomicMaxF(&m[d * 8 + h], lrelu(v));
}

__global__ void k_edge_exp(const int* __restrict__ src, const int* __restrict__ dst,
                           const float* __restrict__ el, const float* __restrict__ er,
                           const float* __restrict__ m, float* __restrict__ denom,
                           float* __restrict__ ez, int E8) {
    int t = blockIdx.x * 256 + threadIdx.x;
    if (t >= E8) return;
    int e = t >> 3, h = t & 7;
    int d = dst[e];
    float v = el[src[e] * 8 + h] + (er ? er[d * 8 + h] : 0.0f);
    float z = expf(lrelu(v) - m[d * 8 + h]);
    ez[t] = z;
    atomicAdd(&denom[d * 8 + h], z);
}

// one thread per (edge, channel): out[dst,c] += (ez/denom) * Z[src,c]
__global__ void k_edge_agg(const int* __restrict__ src, const int* __restrict__ dst,
                           const float* __restrict__ ez, const float* __restrict__ denom,
                           const float* __restrict__ Z, float* __restrict__ out,
                           long long EC) {
    long long t = (long long)blockIdx.x * 256 + threadIdx.x;
    if (t >= EC) return;
    int e = (int)(t >> 7), c = (int)(t & 127);
    int h = c >> 4;
    int s = src[e], d = dst[e];
    float alpha = ez[(long long)e * 8 + h] / denom[d * 8 + h];
    atomicAdd(&out[(long long)d * EMB + c], alpha * Z[(long long)s * EMB + c]);
}

// ---------------------------------------------------------------------------
// Readout layer-1: colsum += sum_rows relu(X @ W + bias)   (WMMA + LDS reduce)
// ---------------------------------------------------------------------------
__global__ void __launch_bounds__(256)
k_mlp_colsum(const float* __restrict__ X, const float* __restrict__ W,
             const float* __restrict__ bias, float* __restrict__ colsum, int numTiles) {
    __shared__ float lsum[EMB];
    if (threadIdx.x < EMB) lsum[threadIdx.x] = 0.0f;
    __syncthreads();

    int wave = threadIdx.x >> 5, lane = threadIdx.x & 31;
    int tile = blockIdx.x * 8 + wave;
    int khalf = lane >> 4, mloc = lane & 15;

    if (tile < numTiles) {
        const float* arow = X + (long long)(tile * 16 + mloc) * EMB;
        v2f a[32];
#pragma unroll
        for (int ks = 0; ks < 32; ++ks) {
            int col = ks * 4 + khalf * 2;
            a[ks][0] = arow[col];
            a[ks][1] = arow[col + 1];
        }
#pragma unroll
        for (int nt = 0; nt < 8; ++nt) {
            int ncol = nt * 16 + mloc;
            v2f b[32];
#pragma unroll
            for (int ks = 0; ks < 32; ++ks) {
                int kk = ks * 4 + khalf * 2;
                b[ks][0] = W[kk * EMB + ncol];
                b[ks][1] = W[(kk + 1) * EMB + ncol];
            }
            v8f c = {};
#pragma unroll
            for (int ks = 0; ks < 32; ++ks) {
                c = __builtin_amdgcn_wmma_f32_16x16x4_f32(
                        false, a[ks], false, b[ks], (short)0, c, false, false);
            }
            float bv = bias[ncol];
            float s = 0.0f;
#pragma unroll
            for (int i = 0; i < 8; ++i) {
                float y = c[i] + bv;
                s += y > 0.0f ? y : 0.0f;
            }
            atomicAdd(&lsum[ncol], s);
        }
    }
    __syncthreads();
    if (threadIdx.x < EMB) atomicAdd(&colsum[threadIdx.x], lsum[threadIdx.x]);
}

// ---------------------------------------------------------------------------
// Final fusion (single block, 128 threads)
// ---------------------------------------------------------------------------
__global__ void __launch_bounds__(128)
k_fuse(const float* __restrict__ cs1, const float* __restrict__ cs2,
       const float* __restrict__ convb,
       const float* __restrict__ rw1, const float* __restrict__ rb1,
       const float* __restrict__ rw2, const float* __restrict__ rb2,
       const float* __restrict__ gran_w, float* __restrict__ out) {
    __shared__ float buf[EMB];
    __shared__ float repr[3][EMB];
    int t = threadIdx.x;

    // gran 1: repr1 = (colsum1/N1) @ rw2[0] + rb2[0]
    buf[t] = cs1[t] * (1.0f / 100000.0f);
    __syncthreads();
    float acc = rb2[0 * EMB + t];
    for (int k = 0; k < EMB; ++k) acc += buf[k] * rw2[0 * EMB * EMB + k * EMB + t];
    repr[0][t] = acc;
    __syncthreads();

    // gran 2
    buf[t] = cs2[t] * (1.0f / 20000.0f);
    __syncthreads();
    acc = rb2[1 * EMB + t];
    for (int k = 0; k < EMB; ++k) acc += buf[k] * rw2[1 * EMB * EMB + k * EMB + t];
    repr[1][t] = acc;
    __syncthreads();

    // gran 3 (constant row: conv_b[2] + conv_b[5])
    buf[t] = convb[2 * EMB + t] + convb[5 * EMB + t];
    __syncthreads();
    acc = rb1[2 * EMB + t];
    for (int k = 0; k < EMB; ++k) acc += buf[k] * rw1[2 * EMB * EMB + k * EMB + t];
    float y = acc > 0.0f ? acc : 0.0f;
    __syncthreads();
    buf[t] = y;
    __syncthreads();
    acc = rb2[2 * EMB + t];
    for (int k = 0; k < EMB; ++k) acc += buf[k] * rw2[2 * EMB * EMB + k * EMB + t];
    repr[2][t] = acc;
    __syncthreads();

    // softmax weights
    float g0 = gran_w[0], g1 = gran_w[1], g2 = gran_w[2];
    float mx = fmaxf(g0, fmaxf(g1, g2));
    float w0 = expf(g0 - mx), w1 = expf(g1 - mx), w2 = expf(g2 - mx);
    float s = w0 + w1 + w2;
    w0 /= s; w1 /= s; w2 /= s;

    out[t]        = w0 * repr[0][t] + w1 * repr[1][t] + w2 * repr[2][t];
    out[128 + t]  = repr[0][t];
    out[256 + t]  = repr[1][t];
    out[384 + t]  = repr[2][t];
    if (t == 0) { out[512] = w0; out[513] = w1; out[514] = w2; }
}

// ---------------------------------------------------------------------------
// launcher
// ---------------------------------------------------------------------------
extern "C" void kernel_launch(void* const* d_in, const int* in_sizes, int n_in,
                              void* d_out, int out_size, void* d_ws, size_t ws_size,
                              hipStream_t stream) {
    const int N1 = 100000, N2 = 20000;
    const int E1 = 600000, EU1 = 200000;

    const int*   item_ids = (const int*)d_in[0];
    const int*   s1s  = (const int*)d_in[1];
    const int*   s1d  = (const int*)d_in[2];
    const int*   u1s  = (const int*)d_in[7];
    const int*   u1d  = (const int*)d_in[8];
    const float* emb  = (const float*)d_in[15];
    const float* fc_w = (const float*)d_in[16];
    const float* al   = (const float*)d_in[17];
    const float* ar   = (const float*)d_in[18];
    const float* cb   = (const float*)d_in[19];
    const float* rw1  = (const float*)d_in[20];
    const float* rb1  = (const float*)d_in[21];
    const float* rw2  = (const float*)d_in[22];
    const float* rb2  = (const float*)d_in[23];
    const float* gw   = (const float*)d_in[24];
    float* out = (float*)d_out;

    float* ws = (float*)d_ws;
    size_t o = 0;
    float* Z0   = ws + o; o += (size_t)N1 * EMB;
    float* Z3   = ws + o; o += (size_t)N1 * EMB;
    float* out1 = ws + o; o += (size_t)N1 * EMB;
    float* out2 = ws + o; o += (size_t)N2 * EMB;
    float* el0  = ws + o; o += (size_t)N1 * 8;
    float* er0  = ws + o; o += (size_t)N1 * 8;
    float* el3  = ws + o; o += (size_t)N1 * 8;
    float* m1   = ws + o; o += (size_t)N1 * 8;
    float* dn1  = ws + o; o += (size_t)N1 * 8;
    float* m2   = ws + o; o += (size_t)N2 * 8;
    float* dn2  = ws + o; o += (size_t)N2 * 8;
    float* ez1  = ws + o; o += (size_t)E1 * 8;
    float* ez2  = ws + o; o += (size_t)EU1 * 8;
    float* cs1  = ws + o; o += 128;
    float* cs2  = ws + o; o += 128;

    const float NEG_INF = -__builtin_huge_valf();
    auto blk = [](long long n) { return (unsigned)((n + 255) / 256); };

    // init
    k_fill<<<blk(N1 * 8), 256, 0, stream>>>(m1, NEG_INF, N1 * 8);
    k_fill<<<blk(N2 * 8), 256, 0, stream>>>(m2, NEG_INF, N2 * 8);
    k_fill<<<blk(N1 * 8), 256, 0, stream>>>(dn1, 0.0f, N1 * 8);
    k_fill<<<blk(N2 * 8), 256, 0, stream>>>(dn2, 0.0f, N2 * 8);
    k_fill<<<1, 256, 0, stream>>>(cs1, 0.0f, 128);
    k_fill<<<1, 256, 0, stream>>>(cs2, 0.0f, 128);
    k_init_out1<<<blk((long long)N1 * EMB), 256, 0, stream>>>(emb, item_ids, cb, out1, N1 * EMB);
    k_init_out2<<<blk((long long)N2 * EMB), 256, 0, stream>>>(cb, out2, N2 * EMB);

    // Z0 = h1@W0, Z3 = h1@W3   (h1 gathered on the fly)
    int tiles1 = N1 / 16; // 6250
    k_gemm_gather_dual<<<(tiles1 + 7) / 8, 256, 0, stream>>>(
        emb, item_ids, fc_w + 0 * EMB * EMB, fc_w + 3 * EMB * EMB, Z0, Z3, tiles1);

    // attention scores
    k_scores<<<blk(N1 * 8), 256, 0, stream>>>(Z0, Z3, al, ar, el0, er0, el3, N1 * 8);

    // edge softmax: seq1 (g0) and up1 (g3, er == 0)
    k_edge_max<<<blk((long long)E1 * 8), 256, 0, stream>>>(s1s, s1d, el0, er0, m1, E1 * 8);
    k_edge_max<<<blk((long long)EU1 * 8), 256, 0, stream>>>(u1s, u1d, el3, nullptr, m2, EU1 * 8);
    k_edge_exp<<<blk((long long)E1 * 8), 256, 0, stream>>>(s1s, s1d, el0, er0, m1, dn1, ez1, E1 * 8);
    k_edge_exp<<<blk((long long)EU1 * 8), 256, 0, stream>>>(u1s, u1d, el3, nullptr, m2, dn2, ez2, EU1 * 8);
    k_edge_agg<<<blk((long long)E1 * EMB), 256, 0, stream>>>(s1s, s1d, ez1, dn1, Z0, out1, (long long)E1 * EMB);
    k_edge_agg<<<blk((long long)EU1 * EMB), 256, 0, stream>>>(u1s, u1d, ez2, dn2, Z3, out2, (long long)EU1 * EMB);

    // readout layer-1 + column sums
    k_mlp_colsum<<<(tiles1 + 7) / 8, 256, 0, stream>>>(out1, rw1 + 0 * EMB * EMB, rb1 + 0 * EMB, cs1, tiles1);
    int tiles2 = N2 / 16; // 1250
    k_mlp_colsum<<<(tiles2 + 7) / 8, 256, 0, stream>>>(out2, rw1 + 1 * EMB * EMB, rb1 + 1 * EMB, cs2, tiles2);

    // final fuse
    k_fuse<<<1, 128, 0, stream>>>(cs1, cs2, cb, rw1, rb1, rw2, rb2, gw, out);
}